// InformerForPointwise_35485019799714
// MI455X (gfx1250) — compile-verified
//
#include <hip/hip_runtime.h>
#include <math.h>

typedef __attribute__((ext_vector_type(2))) float v2f;
typedef __attribute__((ext_vector_type(8))) float v8f;

__device__ __forceinline__ float waveSum(float v) {
#pragma unroll
  for (int off = 16; off > 0; off >>= 1) v += __shfl_xor(v, off, 32);
  return v;
}

// ---------------- WMMA GEMM: C[M,N] = A[M,K] @ W[K,N] (+bias)(gelu)(+res) ----
// one wave -> 64x16 tile (4 row-subtiles), B fragment loaded once per k-step
// and reused by 4 independent WMMAs.
template <int ACT>
__global__ void k_gemm(const float* __restrict__ A, const float* __restrict__ W,
                       const float* __restrict__ bias, const float* __restrict__ res,
                       float* __restrict__ C, int M, int N, int K) {
  int wave = threadIdx.x >> 5, lane = threadIdx.x & 31;
  int half = lane >> 4, l16 = lane & 15;
  int rowBase = (blockIdx.y * 4 + wave) * 64;
  int colBase = blockIdx.x * 16;
  if (rowBase >= M) return;
  v8f acc0 = {}, acc1 = {}, acc2 = {}, acc3 = {};
  const float* Arow0 = A + (size_t)(rowBase + 0 + l16) * K;
  const float* Arow1 = A + (size_t)(rowBase + 16 + l16) * K;
  const float* Arow2 = A + (size_t)(rowBase + 32 + l16) * K;
  const float* Arow3 = A + (size_t)(rowBase + 48 + l16) * K;
  int kOff = 2 * half;
  for (int k = 0; k < K; k += 4) {
    const float* Wr = W + (size_t)(k + kOff) * N + colBase + l16;
    v2f b;
    b.x = Wr[0];
    b.y = Wr[N];
    v2f a0, a1, a2, a3;
    a0.x = Arow0[k + kOff]; a0.y = Arow0[k + kOff + 1];
    a1.x = Arow1[k + kOff]; a1.y = Arow1[k + kOff + 1];
    a2.x = Arow2[k + kOff]; a2.y = Arow2[k + kOff + 1];
    a3.x = Arow3[k + kOff]; a3.y = Arow3[k + kOff + 1];
    acc0 = __builtin_amdgcn_wmma_f32_16x16x4_f32(false, a0, false, b, (short)0, acc0, false, false);
    acc1 = __builtin_amdgcn_wmma_f32_16x16x4_f32(false, a1, false, b, (short)0, acc1, false, false);
    acc2 = __builtin_amdgcn_wmma_f32_16x16x4_f32(false, a2, false, b, (short)0, acc2, false, false);
    acc3 = __builtin_amdgcn_wmma_f32_16x16x4_f32(false, a3, false, b, (short)0, acc3, false, false);
  }
  int col = colBase + l16;
  float bv = bias ? bias[col] : 0.f;
  v8f* accs[4] = {&acc0, &acc1, &acc2, &acc3};
#pragma unroll
  for (int s = 0; s < 4; ++s) {
    v8f& acc = *accs[s];
#pragma unroll
    for (int r = 0; r < 8; ++r) {
      int row = rowBase + s * 16 + r + 8 * half;
      float v = acc[r] + bv;
      if (ACT == 1) v = 0.5f * v * (1.f + erff(v * 0.7071067811865476f));
      if (res) v += res[(size_t)row * N + col];
      C[(size_t)row * N + col] = v;
    }
  }
}

// ------- circular conv1d (k=3, 128->128) via WMMA + BatchNorm + ELU ---------
__global__ void k_conv_bn_elu(const float* __restrict__ x, const float* __restrict__ w,
                              const float* __restrict__ cb, const float* __restrict__ bn_g,
                              const float* __restrict__ bn_b, const float* __restrict__ bn_m,
                              const float* __restrict__ bn_v, float* __restrict__ y,
                              int B, int L) {
  int wave = threadIdx.x >> 5, lane = threadIdx.x & 31;
  int half = lane >> 4, l16 = lane & 15;
  int rowBase = (blockIdx.y * 4 + wave) * 64;
  int colBase = blockIdx.x * 16;
  if (rowBase >= B * L) return;
  int col = colBase + l16;
  int kOff = 2 * half;
  const float* Arow[3][4];
#pragma unroll
  for (int s = 0; s < 4; ++s) {
    int r = rowBase + s * 16 + l16;
    int bidx = r / L, l = r % L;
#pragma unroll
    for (int t = 0; t < 3; ++t) {
      int sl = (l - 1 + t + L) % L;
      Arow[t][s] = x + ((size_t)bidx * L + sl) * 128;
    }
  }
  v8f acc0 = {}, acc1 = {}, acc2 = {}, acc3 = {};
  for (int t = 0; t < 3; ++t) {
    const float* Wc = w + (size_t)col * 384 + t;
    for (int k = 0; k < 128; k += 4) {
      v2f b;
      b.x = Wc[(k + kOff) * 3];
      b.y = Wc[(k + kOff + 1) * 3];
      v2f a0, a1, a2, a3;
      a0.x = Arow[t][0][k + kOff]; a0.y = Arow[t][0][k + kOff + 1];
      a1.x = Arow[t][1][k + kOff]; a1.y = Arow[t][1][k + kOff + 1];
      a2.x = Arow[t][2][k + kOff]; a2.y = Arow[t][2][k + kOff + 1];
      a3.x = Arow[t][3][k + kOff]; a3.y = Arow[t][3][k + kOff + 1];
      acc0 = __builtin_amdgcn_wmma_f32_16x16x4_f32(false, a0, false, b, (short)0, acc0, false, false);
      acc1 = __builtin_amdgcn_wmma_f32_16x16x4_f32(false, a1, false, b, (short)0, acc1, false, false);
      acc2 = __builtin_amdgcn_wmma_f32_16x16x4_f32(false, a2, false, b, (short)0, acc2, false, false);
      acc3 = __builtin_amdgcn_wmma_f32_16x16x4_f32(false, a3, false, b, (short)0, acc3, false, false);
    }
  }
  float cbv = cb[col];
  float inv = rsqrtf(bn_v[col] + 1e-5f);
  float g = bn_g[col], bb = bn_b[col], m = bn_m[col];
  v8f* accs[4] = {&acc0, &acc1, &acc2, &acc3};
#pragma unroll
  for (int s = 0; s < 4; ++s) {
    v8f& acc = *accs[s];
#pragma unroll
    for (int rr = 0; rr < 8; ++rr) {
      int row = rowBase + s * 16 + rr + 8 * half;
      float val = acc[rr] + cbv;
      val = (val - m) * inv * g + bb;
      val = val > 0.f ? val : expm1f(val);
      y[(size_t)row * 128 + col] = val;
    }
  }
}

// ---------------- maxpool k=3 s=2, -inf pad -------------------------------
__global__ void k_maxpool(const float* __restrict__ x, float* __restrict__ y, int B,
                          int Lin, int Lout) {
  int i = blockIdx.x * blockDim.x + threadIdx.x;
  int total = B * Lout * 128;
  if (i >= total) return;
  int d = i & 127, r = i >> 7;
  int lo = r % Lout, b = r / Lout;
  float m = -1e30f;
  for (int t = 0; t < 3; ++t) {
    int l = 2 * lo - 1 + t;
    if (l >= 0 && l < Lin) m = fmaxf(m, x[((size_t)b * Lin + l) * 128 + d]);
  }
  y[i] = m;
}

// ---------------- layernorm over last dim (128), wave per row --------------
__global__ void k_ln(const float* __restrict__ x, const float* __restrict__ g,
                     const float* __restrict__ b, float* __restrict__ y, int rows) {
  int row = blockIdx.x * (blockDim.x >> 5) + (threadIdx.x >> 5);
  int lane = threadIdx.x & 31;
  if (row >= rows) return;
  const float* xr = x + (size_t)row * 128;
  float v[4], s = 0.f;
#pragma unroll
  for (int i = 0; i < 4; ++i) { v[i] = xr[lane + 32 * i]; s += v[i]; }
  s = waveSum(s);
  float mu = s * (1.f / 128.f);
  float q = 0.f;
#pragma unroll
  for (int i = 0; i < 4; ++i) { float d = v[i] - mu; q += d * d; }
  q = waveSum(q);
  float inv = rsqrtf(q * (1.f / 128.f) + 1e-5f);
  float* yr = y + (size_t)row * 128;
#pragma unroll
  for (int i = 0; i < 4; ++i) {
    int d = lane + 32 * i;
    yr[d] = (v[i] - mu) * inv * g[d] + b[d];
  }
}

// ---------------- embeddings ----------------------------------------------
__device__ __forceinline__ float posemb(int l, int d) {
  int hd = d >> 1;
  float freq = __expf(-(float)(2 * hd) * (9.210340371976184f / 128.f));
  float ang = (float)l * freq;
  float p = (d & 1) ? cosf(ang) : sinf(ang);
  p += (d & 1) ? 5.f : 0.f;  // TEMPORAL0: sum of 5 cos(0) on odd dims
  return p;
}

__global__ void k_embed(const float* __restrict__ x, const float* __restrict__ w,
                        const float* __restrict__ cb, float* __restrict__ out, int B,
                        int L, int Cin) {
  int i = blockIdx.x * blockDim.x + threadIdx.x;
  int total = B * L * 128;
  if (i >= total) return;
  int d = i & 127, bl = i >> 7;
  int l = bl % L, bb = bl / L;
  float acc = cb[d];
  for (int t = 0; t < 3; ++t) {
    int sl = (l - 1 + t + L) % L;
    const float* xr = x + ((size_t)bb * L + sl) * Cin;
    const float* wr = w + (size_t)d * Cin * 3 + t;
    for (int ci = 0; ci < Cin; ++ci) acc += xr[ci] * wr[ci * 3];
  }
  out[i] = acc + posemb(l, d);
}

__global__ void k_embed_zero(const float* __restrict__ cb, float* __restrict__ out,
                             int B, int L) {
  int i = blockIdx.x * blockDim.x + threadIdx.x;
  int total = B * L * 128;
  if (i >= total) return;
  int d = i & 127, l = (i >> 7) % L;
  out[i] = cb[d] + posemb(l, d);
}

// ---------------- ProbSparse attention pieces ------------------------------
__global__ void k_fill_idx(int* __restrict__ idx, int n, int LK, unsigned seed) {
  int i = blockIdx.x * blockDim.x + threadIdx.x;
  if (i >= n) return;
  unsigned x = (unsigned)i * 2654435761u ^ seed;
  x ^= x >> 16; x *= 0x85EBCA6Bu; x ^= x >> 13; x *= 0xC2B2AE35u; x ^= x >> 16;
  idx[i] = (int)(x % (unsigned)LK);
}

__global__ void k_prob_M(const float* __restrict__ Q, const float* __restrict__ K,
                         const int* __restrict__ idx, float* __restrict__ Mout, int B,
                         int H, int LQ, int LK, int U) {
  int wid = blockIdx.x * (blockDim.x >> 5) + (threadIdx.x >> 5);
  if (wid >= B * H * LQ) return;
  int lane = threadIdx.x & 31;
  int l = wid % LQ, bh = wid / LQ;
  int h = bh % H, b = bh / H;
  float qd = Q[((size_t)b * LQ + l) * 128 + h * 32 + lane];
  float mx = -1e30f, sm = 0.f;
  for (int j = 0; j < U; ++j) {
    int k = idx[l * U + j];
    float s = waveSum(qd * K[((size_t)b * LK + k) * 128 + h * 32 + lane]);
    mx = fmaxf(mx, s);
    sm += s;
  }
  if (lane == 0) Mout[wid] = mx - sm / (float)U;
}

__global__ void k_topk(const float* __restrict__ M, int* __restrict__ topidx, int LQ,
                       int u) {
  __shared__ unsigned char used[256];
  int bh = blockIdx.x, lane = threadIdx.x;
  for (int i = lane; i < LQ; i += 32) used[i] = 0;
  __syncthreads();
  const float* Mr = M + (size_t)bh * LQ;
  for (int it = 0; it < u; ++it) {
    float best = -1e30f;
    int bidx = 1 << 30;
    for (int l = lane; l < LQ; l += 32) {
      if (!used[l]) {
        float v = Mr[l];
        if (v > best || (v == best && l < bidx)) { best = v; bidx = l; }
      }
    }
    for (int off = 16; off > 0; off >>= 1) {
      float ov = __shfl_xor(best, off, 32);
      int oi = __shfl_xor(bidx, off, 32);
      if (ov > best || (ov == best && oi < bidx)) { best = ov; bidx = oi; }
    }
    if (lane == 0) { topidx[(size_t)bh * u + it] = bidx; used[bidx] = 1; }
    __syncthreads();
  }
}

__global__ void k_ctx_mean(const float* __restrict__ V, float* __restrict__ ctx, int B,
                           int H, int LQ, int LK) {
  int i = blockIdx.x * blockDim.x + threadIdx.x;
  if (i >= B * H * 32) return;
  int d = i & 31, bh = i >> 5;
  int h = bh % H, b = bh / H;
  float s = 0.f;
  for (int k = 0; k < LK; ++k) s += V[((size_t)b * LK + k) * 128 + h * 32 + d];
  s /= (float)LK;
  float* cr = ctx + (size_t)bh * LQ * 32 + d;
  for (int l = 0; l < LQ; ++l) cr[(size_t)l * 32] = s;
}

__global__ void k_ctx_cumsum(const float* __restrict__ V, float* __restrict__ ctx, int B,
                             int H, int L) {
  int i = blockIdx.x * blockDim.x + threadIdx.x;
  if (i >= B * H * 32) return;
  int d = i & 31, bh = i >> 5;
  int h = bh % H, b = bh / H;
  float s = 0.f;
  for (int l = 0; l < L; ++l) {
    s += V[((size_t)b * L + l) * 128 + h * 32 + d];
    ctx[((size_t)bh * L + l) * 32 + d] = s;
  }
}

__global__ void k_prob_update(const float* __restrict__ Q, const float* __restrict__ K,
                              const float* __restrict__ V, const int* __restrict__ topidx,
                              float* __restrict__ ctx, int B, int H, int LQ, int LK,
                              int u, int causal) {
  int wid = blockIdx.x * (blockDim.x >> 5) + (threadIdx.x >> 5);
  if (wid >= B * H * u) return;
  int lane = threadIdx.x & 31;
  int ui = wid % u, bh = wid / u;
  int h = bh % H, b = bh / H;
  int row = topidx[(size_t)bh * u + ui];
  float qd = Q[((size_t)b * LQ + row) * 128 + h * 32 + lane] * 0.17677669529663687f;
  int kend = causal ? (row + 1) : LK;
  float m = -1e30f, lsum = 0.f, acc = 0.f;
  for (int k = 0; k < kend; ++k) {
    float s = waveSum(qd * K[((size_t)b * LK + k) * 128 + h * 32 + lane]);
    float vd = V[((size_t)b * LK + k) * 128 + h * 32 + lane];
    float mn = fmaxf(m, s);
    float c = __expf(m - mn), p = __expf(s - mn);
    lsum = lsum * c + p;
    acc = acc * c + p * vd;
    m = mn;
  }
  ctx[((size_t)bh * LQ + row) * 32 + lane] = acc / lsum;
}

__global__ void k_full_attn(const float* __restrict__ Q, const float* __restrict__ K,
                            const float* __restrict__ V, float* __restrict__ ctx, int B,
                            int H, int LQ, int LK) {
  int wid = blockIdx.x * (blockDim.x >> 5) + (threadIdx.x >> 5);
  if (wid >= B * H * LQ) return;
  int lane = threadIdx.x & 31;
  int l = wid % LQ, bh = wid / LQ;
  int h = bh % H, b = bh / H;
  float qd = Q[((size_t)b * LQ + l) * 128 + h * 32 + lane] * 0.17677669529663687f;
  float m = -1e30f, lsum = 0.f, acc = 0.f;
  for (int k = 0; k < LK; ++k) {
    float s = waveSum(qd * K[((size_t)b * LK + k) * 128 + h * 32 + lane]);
    float vd = V[((size_t)b * LK + k) * 128 + h * 32 + lane];
    float mn = fmaxf(m, s);
    float c = __expf(m - mn), p = __expf(s - mn);
    lsum = lsum * c + p;
    acc = acc * c + p * vd;
    m = mn;
  }
  ctx[((size_t)bh * LQ + l) * 32 + lane] = acc / lsum;
}

__global__ void k_ctx_to_out(const float* __restrict__ ctx, float* __restrict__ out,
                             int B, int H, int LQ, int mix) {
  int i = blockIdx.x * blockDim.x + threadIdx.x;
  if (i >= B * H * LQ * 32) return;
  int d = i & 31, rest = i >> 5;
  int l = rest % LQ, bh = rest / LQ;
  int h = bh % H, b = bh / H;
  float v = ctx[i];
  if (mix) {
    int p = (h * LQ + l) * 32 + d;  // reshape(B,H,L,Dh) -> (B,L,128)
    out[(size_t)b * LQ * 128 + p] = v;
  } else {
    out[((size_t)b * LQ + l) * 128 + h * 32 + d] = v;
  }
}

__global__ void k_proj(const float* __restrict__ x, const float* __restrict__ w,
                       const float* __restrict__ b, float* __restrict__ out, int rows) {
  int i = blockIdx.x * blockDim.x + threadIdx.x;
  if (i >= rows * 6) return;
  int c = i % 6, r = i / 6;
  const float* xr = x + (size_t)r * 128;
  float acc = b[c];
  for (int k = 0; k < 128; ++k) acc += xr[k] * w[k * 6 + c];
  out[i] = acc;
}

// ===========================================================================
extern "C" void kernel_launch(void* const* d_in, const int* in_sizes, int n_in,
                              void* d_out, int out_size, void* d_ws, size_t ws_size,
                              hipStream_t stream) {
  (void)in_sizes; (void)n_in; (void)out_size; (void)ws_size;
  const int B = 64, H = 4, DM = 128, DFF = 512;
  auto div_up = [](int a, int b) { return (a + b - 1) / b; };

  // ---- decode params (JAX pytree flatten: dicts in sorted key order) ----
  int pi = 0;
  auto F = [&]() { return (const float*)d_in[pi++]; };
  const float* x_enc = F();
  struct ConvBlk { const float *bn_b, *bn_g, *bn_m, *bn_v, *cb, *cw; } convs[2];
  for (int i = 0; i < 2; ++i) {
    convs[i].bn_b = F(); convs[i].bn_g = F(); convs[i].bn_m = F();
    convs[i].bn_v = F(); convs[i].cb = F(); convs[i].cw = F();
  }
  struct LinW { const float *b, *w; };
  auto rdLin = [&]() { LinW l; l.b = F(); l.w = F(); return l; };
  struct AttnW { LinW k, o, q, v; };
  auto rdAttn = [&]() { AttnW a; a.k = rdLin(); a.o = rdLin(); a.q = rdLin(); a.v = rdLin(); return a; };
  struct LNW { const float *b, *g; };
  auto rdLN = [&]() { LNW n; n.b = F(); n.g = F(); return n; };
  struct DecL { AttnW cross; LinW ff1, ff2; LNW n1, n2, n3; AttnW self; } dec[2];
  for (int j = 0; j < 2; ++j) {
    dec[j].cross = rdAttn(); dec[j].ff1 = rdLin(); dec[j].ff2 = rdLin();
    dec[j].n1 = rdLN(); dec[j].n2 = rdLN(); dec[j].n3 = rdLN(); dec[j].self = rdAttn();
  }
  LNW dec_norm = rdLN();
  LinW dec_tok = rdLin();
  struct EncL { AttnW attn; LinW ff1, ff2; LNW n1, n2; } enc[3];
  for (int i = 0; i < 3; ++i) {
    enc[i].attn = rdAttn(); enc[i].ff1 = rdLin(); enc[i].ff2 = rdLin();
    enc[i].n1 = rdLN(); enc[i].n2 = rdLN();
  }
  LNW enc_norm = rdLN();
  LinW enc_tok = rdLin();
  LinW proj = rdLin();

  // ---- workspace arena ----
  float* ws = (float*)d_ws;
  size_t off = 0;
  auto alloc = [&](size_t n) { float* p = ws + off; off += n; return p; };
  const size_t ACT = (size_t)B * 200 * 128;
  float* bufX = alloc(ACT);
  float* bufU = alloc(ACT);
  float* bufT = alloc(ACT);
  float* bufQ = alloc(ACT);
  float* bufK = alloc(ACT);
  float* bufV = alloc(ACT);
  float* bufCtx = alloc(ACT);
  float* bufFF = alloc((size_t)B * 200 * 512);
  float* encOut = alloc((size_t)B * 50 * 128);
  float* Mbuf = alloc((size_t)B * H * 200);
  int* topidx = (int*)alloc((size_t)B * H * 32);
  int* idxbuf = (int*)alloc((size_t)200 * 32);

  auto gemm = [&](const float* A, const LinW& L, float* C, int M, int N, int K,
                  const float* res, int act) {
    dim3 g(N / 16, div_up(M, 256)), blk(128);
    if (act)
      k_gemm<1><<<g, blk, 0, stream>>>(A, L.w, L.b, res, C, M, N, K);
    else
      k_gemm<0><<<g, blk, 0, stream>>>(A, L.w, L.b, res, C, M, N, K);
  };
  auto lnorm = [&](const float* x, const LNW& n, float* y, int rows) {
    k_ln<<<div_up(rows, 8), 256, 0, stream>>>(x, n.g, n.b, y, rows);
  };
  auto attn = [&](const AttnW& w, const float* qin, const float* kvin, int Lq, int Lk,
                  bool prob, bool mask, bool mix, int seed, float* outbuf) {
    gemm(qin, w.q, bufQ, B * Lq, DM, DM, nullptr, 0);
    gemm(kvin, w.k, bufK, B * Lk, DM, DM, nullptr, 0);
    gemm(kvin, w.v, bufV, B * Lk, DM, DM, nullptr, 0);
    if (prob) {
      int U = 5 * (int)ceil(log((double)Lk)); if (U > Lk) U = Lk;
      int u = 5 * (int)ceil(log((double)Lq)); if (u > Lq) u = Lq;
      k_fill_idx<<<div_up(Lq * U, 256), 256, 0, stream>>>(
          idxbuf, Lq * U, Lk, (unsigned)(seed * 2654435761u + 12345u));
      k_prob_M<<<div_up(B * H * Lq, 4), 128, 0, stream>>>(bufQ, bufK, idxbuf, Mbuf, B,
                                                          H, Lq, Lk, U);
      k_topk<<<B * H, 32, 0, stream>>>(Mbuf, topidx, Lq, u);
      if (mask)
        k_ctx_cumsum<<<div_up(B * H * 32, 256), 256, 0, stream>>>(bufV, bufCtx, B, H, Lq);
      else
        k_ctx_mean<<<div_up(B * H * 32, 256), 256, 0, stream>>>(bufV, bufCtx, B, H, Lq, Lk);
      k_prob_update<<<div_up(B * H * u, 4), 128, 0, stream>>>(
          bufQ, bufK, bufV, topidx, bufCtx, B, H, Lq, Lk, u, mask ? 1 : 0);
    } else {
      k_full_attn<<<div_up(B * H * Lq, 4), 128, 0, stream>>>(bufQ, bufK, bufV, bufCtx,
                                                             B, H, Lq, Lk);
    }
    k_ctx_to_out<<<div_up(B * H * Lq * 32, 256), 256, 0, stream>>>(bufCtx, bufT, B, H,
                                                                   Lq, mix ? 1 : 0);
    gemm(bufT, w.o, outbuf, B * Lq, DM, DM, qin, 0);  // + residual (x + attn)
  };

  // ================= encoder =================
  k_embed<<<div_up(B * 200 * DM, 256), 256, 0, stream>>>(x_enc, enc_tok.w, enc_tok.b,
                                                         bufX, B, 200, 12);
  int L = 200;
  for (int i = 0; i < 3; ++i) {
    attn(enc[i].attn, bufX, bufX, L, L, true, false, false, i, bufU);
    lnorm(bufU, enc[i].n1, bufX, B * L);
    gemm(bufX, enc[i].ff1, bufFF, B * L, DFF, DM, nullptr, 1);   // gelu
    gemm(bufFF, enc[i].ff2, bufU, B * L, DM, DFF, bufX, 0);      // + residual
    lnorm(bufU, enc[i].n2, bufX, B * L);
    if (i < 2) {
      dim3 g(DM / 16, div_up(B * L, 256)), blk(128);
      k_conv_bn_elu<<<g, blk, 0, stream>>>(bufX, convs[i].cw, convs[i].cb, convs[i].bn_g,
                                           convs[i].bn_b, convs[i].bn_m, convs[i].bn_v,
                                           bufU, B, L);
      int Lout = (L - 1) / 2 + 1;
      k_maxpool<<<div_up(B * Lout * DM, 256), 256, 0, stream>>>(bufU, bufX, B, L, Lout);
      L = Lout;
    }
  }
  lnorm(bufX, enc_norm, encOut, B * L);  // L == 50

  // ================= decoder =================
  k_embed_zero<<<div_up(B * 200 * DM, 256), 256, 0, stream>>>(dec_tok.b, bufX, B, 200);
  for (int j = 0; j < 2; ++j) {
    attn(dec[j].self, bufX, bufX, 200, 200, true, true, true, 3 + j, bufU);
    lnorm(bufU, dec[j].n1, bufX, B * 200);
    attn(dec[j].cross, bufX, encOut, 200, L, false, false, false, 0, bufU);
    lnorm(bufU, dec[j].n2, bufX, B * 200);
    gemm(bufX, dec[j].ff1, bufFF, B * 200, DFF, DM, nullptr, 1);
    gemm(bufFF, dec[j].ff2, bufU, B * 200, DM, DFF, bufX, 0);
    lnorm(bufU, dec[j].n3, bufX, B * 200);
  }
  lnorm(bufX, dec_norm, bufU, B * 200);
  k_proj<<<div_up(B * 200 * 6, 256), 256, 0, stream>>>(bufU, proj.w, proj.b,
                                                       (float*)d_out, B * 200);
}